// GShardMoELayer_2216203125409
// MI455X (gfx1250) — compile-verified
//
#include <hip/hip_runtime.h>

#define S_TOK 4096
#define M_DIM 1024
#define E_NUM 8
#define CAP   1024

typedef float v2f __attribute__((ext_vector_type(2)));
typedef float v8f __attribute__((ext_vector_type(8)));

// ---------------------------------------------------------------------------
// K1: gating — one wave per token: logits = x[s,:] @ wg, softmax, top1/top2
// ---------------------------------------------------------------------------
__global__ __launch_bounds__(256) void moe_gate_kernel(
    const float* __restrict__ x, const float* __restrict__ wg,
    float* __restrict__ gates_out, int* __restrict__ idxp_out)
{
  int s    = blockIdx.x * 8 + (threadIdx.x >> 5);
  int lane = threadIdx.x & 31;
  float acc[8];
  #pragma unroll
  for (int e = 0; e < 8; ++e) acc[e] = 0.f;
  const float* xs = x + (size_t)s * M_DIM;
  for (int m = lane; m < M_DIM; m += 32) {
    float xv = xs[m];
    const float* wr = wg + m * 8;
    #pragma unroll
    for (int e = 0; e < 8; ++e) acc[e] = fmaf(xv, wr[e], acc[e]);
  }
  #pragma unroll
  for (int off = 16; off >= 1; off >>= 1) {
    #pragma unroll
    for (int e = 0; e < 8; ++e) acc[e] += __shfl_xor(acc[e], off, 32);
  }
  if (lane == 0) {
    float mx = acc[0];
    #pragma unroll
    for (int e = 1; e < 8; ++e) mx = fmaxf(mx, acc[e]);
    float g[8]; float se = 0.f;
    #pragma unroll
    for (int e = 0; e < 8; ++e) { g[e] = expf(acc[e] - mx); se += g[e]; }
    float inv = 1.f / se;
    int i1 = 0; float b1 = acc[0];
    #pragma unroll
    for (int e = 1; e < 8; ++e) if (acc[e] > b1) { b1 = acc[e]; i1 = e; }
    int i2 = (i1 == 0) ? 1 : 0; float b2 = acc[i2];
    #pragma unroll
    for (int e = 0; e < 8; ++e)
      if (e != i1 && acc[e] > b2) { b2 = acc[e]; i2 = e; }
    float* go = gates_out + (size_t)s * 8;
    #pragma unroll
    for (int e = 0; e < 8; ++e) go[e] = g[e] * inv;
    idxp_out[s] = i1 | (i2 << 8);
  }
}

// ---------------------------------------------------------------------------
// K2: single-wave assignment — ballot-based stable per-expert ranking,
//     capacity drop, weight renorm, slot2tok scatter, l_aux
// ---------------------------------------------------------------------------
__global__ __launch_bounds__(32) void moe_assign_kernel(
    const float* __restrict__ gates, const int* __restrict__ idxp,
    int* __restrict__ slot2tok, int* __restrict__ meta,
    float* __restrict__ wgt, float* __restrict__ laux_out)
{
  __shared__ int loc1_s[S_TOK];
  int lane = threadIdx.x;
  for (int i = lane; i < E_NUM * CAP; i += 32) slot2tok[i] = -1;

  unsigned below = (1u << lane) - 1u;
  int cnt1[8];
  #pragma unroll
  for (int e = 0; e < 8; ++e) cnt1[e] = 0;
  for (int base = 0; base < S_TOK; base += 32) {
    int e1 = idxp[base + lane] & 0xff;
    int myloc = 0;
    #pragma unroll
    for (int e = 0; e < 8; ++e) {
      unsigned m = (unsigned)__ballot(e1 == e);
      if (e1 == e) myloc = cnt1[e] + __popc(m & below);
      cnt1[e] += __popc(m);
    }
    loc1_s[base + lane] = myloc;
  }
  __syncthreads();
  // cnt1[] now holds pre-drop totals of mask1 per expert (all lanes uniform)
  int cnt2[8];
  #pragma unroll
  for (int e = 0; e < 8; ++e) cnt2[e] = 0;
  for (int base = 0; base < S_TOK; base += 32) {
    int sidx = base + lane;
    int p  = idxp[sidx];
    int e1 = p & 0xff, e2 = (p >> 8) & 0xff;
    int myloc2 = 0;
    #pragma unroll
    for (int e = 0; e < 8; ++e) {
      unsigned m = (unsigned)__ballot(e2 == e);
      if (e2 == e) myloc2 = cnt2[e] + __popc(m & below);
      cnt2[e] += __popc(m);
    }
    int l2 = myloc2 + cnt1[e2];
    int l1 = loc1_s[sidx];
    float g1 = gates[(size_t)sidx * 8 + e1];
    float g2 = gates[(size_t)sidx * 8 + e2];
    bool k1 = (l1 < CAP), k2 = (l2 < CAP);
    float g1m = k1 ? g1 : 0.f, g2m = k2 ? g2 : 0.f;
    float den = fmaxf(g1m + g2m, 1.1920929e-07f);
    wgt[2 * sidx]     = g1m / den;
    wgt[2 * sidx + 1] = g2m / den;
    meta[2 * sidx]     = k1 ? ((e1 << 16) | l1) : -1;
    meta[2 * sidx + 1] = k2 ? ((e2 << 16) | l2) : -1;
    if (k1) slot2tok[e1 * CAP + l1] = sidx;
    if (k2) slot2tok[e2 * CAP + l2] = sidx;
  }
  // l_aux = mean(me*ce) * E*E   (ce from pre-drop mask1 counts)
  float gs[8];
  #pragma unroll
  for (int e = 0; e < 8; ++e) gs[e] = 0.f;
  for (int sidx = lane; sidx < S_TOK; sidx += 32) {
    const float* gr = gates + (size_t)sidx * 8;
    #pragma unroll
    for (int e = 0; e < 8; ++e) gs[e] += gr[e];
  }
  #pragma unroll
  for (int off = 16; off >= 1; off >>= 1) {
    #pragma unroll
    for (int e = 0; e < 8; ++e) gs[e] += __shfl_xor(gs[e], off, 32);
  }
  if (lane == 0) {
    float la = 0.f;
    #pragma unroll
    for (int e = 0; e < 8; ++e)
      la += (gs[e] / (float)S_TOK) * ((float)cnt1[e] / (float)S_TOK);
    laux_out[0] = la * 8.f;   // (la/8) * 64
  }
}

// ---------------------------------------------------------------------------
// K3: expert GEMM, fp32 WMMA 16x16x4, dispatch-gather fused into A load.
//     Block = 256 thr (8 waves) -> 128x128 tile; K staged by 32 through LDS.
// ---------------------------------------------------------------------------
#define TILE 128
#define KB   32
#define LSTR 36   // KB + 4 padding: conflict-free, keeps float2/float4 alignment

__global__ __launch_bounds__(256) void moe_expert_gemm_kernel(
    const float* __restrict__ x, const float* __restrict__ ew,
    const float* __restrict__ eb, const int* __restrict__ slot2tok,
    float* __restrict__ eo)
{
  __shared__ float Al[TILE * LSTR];   // A rows [row][k]
  __shared__ float Bt[TILE * LSTR];   // B transposed [n][k]
  int e       = blockIdx.z;
  int rowBase = blockIdx.y * TILE;
  int colBase = blockIdx.x * TILE;
  int tid  = threadIdx.x;
  int lane = tid & 31;
  int waveRow = (tid >> 5) * 16;
  int half = lane >> 4;       // K-offset selector (0 -> K,K+1 ; 1 -> K+2,K+3)
  int nloc = lane & 15;

  int tokr[4], arow4[4], aseg[4];
  #pragma unroll
  for (int i = 0; i < 4; ++i) {
    int idx = tid + i * 256;
    arow4[i] = idx >> 3;
    aseg[i]  = idx & 7;
    tokr[i]  = slot2tok[e * CAP + rowBase + arow4[i]];
  }

  v8f acc[8];
  #pragma unroll
  for (int j = 0; j < 8; ++j)
    #pragma unroll
    for (int v = 0; v < 8; ++v) acc[j][v] = 0.f;

  const float* W = ew + (size_t)e * M_DIM * M_DIM;

  for (int k0 = 0; k0 < M_DIM; k0 += KB) {
    // stage A (gathered token rows; empty slot -> zeros)
    #pragma unroll
    for (int i = 0; i < 4; ++i) {
      float4 v = make_float4(0.f, 0.f, 0.f, 0.f);
      if (tokr[i] >= 0)
        v = *(const float4*)(x + (size_t)tokr[i] * M_DIM + k0 + aseg[i] * 4);
      *(float4*)(&Al[arow4[i] * LSTR + aseg[i] * 4]) = v;
    }
    // stage B transposed: W[k][n] -> Bt[n][k]
    #pragma unroll
    for (int i = 0; i < 4; ++i) {
      int idx = tid + i * 256;
      int kr  = idx >> 5;
      int seg = idx & 31;
      float4 v = *(const float4*)(W + (size_t)(k0 + kr) * M_DIM + colBase + seg * 4);
      Bt[(seg * 4 + 0) * LSTR + kr] = v.x;
      Bt[(seg * 4 + 1) * LSTR + kr] = v.y;
      Bt[(seg * 4 + 2) * LSTR + kr] = v.z;
      Bt[(seg * 4 + 3) * LSTR + kr] = v.w;
    }
    __syncthreads();
    #pragma unroll
    for (int kk = 0; kk < KB; kk += 4) {
      v2f af = *(const v2f*)(&Al[(waveRow + nloc) * LSTR + kk + half * 2]);
      #pragma unroll
      for (int j = 0; j < 8; ++j) {
        v2f bf = *(const v2f*)(&Bt[(j * 16 + nloc) * LSTR + kk + half * 2]);
        acc[j] = __builtin_amdgcn_wmma_f32_16x16x4_f32(
            false, af, false, bf, (short)0, acc[j], false, false);
      }
    }
    __syncthreads();
  }
  // epilogue: + bias, store expert_out[e][row][col]
  #pragma unroll
  for (int j = 0; j < 8; ++j) {
    int col = colBase + j * 16 + nloc;
    float bias = eb[e * M_DIM + col];
    #pragma unroll
    for (int v = 0; v < 8; ++v) {
      int row = rowBase + waveRow + v + half * 8;
      eo[((size_t)e * CAP + row) * M_DIM + col] = acc[j][v] + bias;
    }
  }
}

// ---------------------------------------------------------------------------
// K4: combine — out[s,:] = w1*EO[e1,c1,:] + w2*EO[e2,c2,:]
// ---------------------------------------------------------------------------
__global__ __launch_bounds__(256) void moe_combine_kernel(
    const float* __restrict__ eo, const int* __restrict__ meta,
    const float* __restrict__ wgt, float* __restrict__ out)
{
  int s  = blockIdx.x;
  int m1 = meta[2 * s], m2 = meta[2 * s + 1];
  float w1 = wgt[2 * s], w2 = wgt[2 * s + 1];
  int m4 = threadIdx.x * 4;
  float4 a = make_float4(0.f, 0.f, 0.f, 0.f);
  if (m1 >= 0) {
    float4 v = *(const float4*)(eo + ((size_t)(m1 >> 16) * CAP + (m1 & 0xffff)) * M_DIM + m4);
    a.x += w1 * v.x; a.y += w1 * v.y; a.z += w1 * v.z; a.w += w1 * v.w;
  }
  if (m2 >= 0) {
    float4 v = *(const float4*)(eo + ((size_t)(m2 >> 16) * CAP + (m2 & 0xffff)) * M_DIM + m4);
    a.x += w2 * v.x; a.y += w2 * v.y; a.z += w2 * v.z; a.w += w2 * v.w;
  }
  *(float4*)(out + (size_t)s * M_DIM + m4) = a;
}

// ---------------------------------------------------------------------------
extern "C" void kernel_launch(void* const* d_in, const int* in_sizes, int n_in,
                              void* d_out, int out_size, void* d_ws, size_t ws_size,
                              hipStream_t stream) {
  (void)in_sizes; (void)n_in; (void)out_size; (void)ws_size;
  const float* x  = (const float*)d_in[0];   // [2,2048,1024]
  const float* wg = (const float*)d_in[1];   // [1024,8]
  const float* ew = (const float*)d_in[2];   // [8,1024,1024]
  const float* eb = (const float*)d_in[3];   // [8,1024]
  float* out = (float*)d_out;                // [4096*1024] ++ [1] l_aux

  char* ws = (char*)d_ws;
  float* gates    = (float*)(ws);            //  131072 B : [S,8]
  int*   idxp     = (int*)  (ws + 131072);   //   16384 B : [S]
  int*   slot2tok = (int*)  (ws + 147456);   //   32768 B : [E,CAP]
  int*   meta     = (int*)  (ws + 180224);   //   32768 B : [S,2]
  float* wgt      = (float*)(ws + 212992);   //   32768 B : [S,2]
  float* eo       = (float*)(ws + 245760);   // 32 MB     : [E,CAP,M]

  moe_gate_kernel<<<S_TOK / 8, 256, 0, stream>>>(x, wg, gates, idxp);
  moe_assign_kernel<<<1, 32, 0, stream>>>(gates, idxp, slot2tok, meta, wgt,
                                          out + (size_t)S_TOK * M_DIM);
  moe_expert_gemm_kernel<<<dim3(M_DIM / TILE, CAP / TILE, E_NUM), 256, 0, stream>>>(
      x, ew, eb, slot2tok, eo);
  moe_combine_kernel<<<S_TOK, 256, 0, stream>>>(eo, meta, wgt, out);
}